// DeformableConv2d_46574625357923
// MI455X (gfx1250) — compile-verified
//
#include <hip/hip_runtime.h>
#include <hip/hip_bf16.h>
#include <math.h>

// ---------------------------------------------------------------------------
// Deformable Conv2d (torchvision v1 semantics), stride=1, pad=1, 3x3.
//   x        [8,64,128,128]  f32
//   w_offset [18,64,3,3]     f32   (offset-predicting conv, no bias)
//   w_deform [64,64,3,3]     f32
//   b_deform [64]            f32
//   out      [8,64,128,128]  f32
//
// Strategy: both convs are GEMMs run on V_WMMA_F32_16X16X4_F32 (fp32 WMMA,
// exact fp32 math like the reference). B-matrix (im2col / bilinear-sampled
// values) is built on the fly per lane. x is pre-transposed to channels-last
// so one global_load_b64 fetches both channels a lane owns per K-step.
// ---------------------------------------------------------------------------

#define BATCH 8
#define CH    64
#define HH    128
#define WW    128
#define OCH   64
#define KTAP  9     // 3x3

typedef float v2f __attribute__((ext_vector_type(2)));
typedef float v8f __attribute__((ext_vector_type(8)));

#define WMMA_F32(A, B, C) \
    __builtin_amdgcn_wmma_f32_16x16x4_f32(false, (A), false, (B), (short)0, (C), false, false)

// --------------------------------------------------------------------------
// Kernel 1: transpose x [B][C][H][W] -> xT [B][H][W][C]  (channels-last)
// One block per (b,y) row, LDS staged to keep both phases coalesced.
// --------------------------------------------------------------------------
__global__ __launch_bounds__(256) void k_transpose_x(const float* __restrict__ x,
                                                     float* __restrict__ xT) {
    __shared__ float tile[64 * 129];  // pad stride 129 -> bank-conflict free
    const int by = blockIdx.x;        // 0..B*H-1
    const int b = by >> 7;
    const int y = by & 127;
    const int tid = threadIdx.x;

    for (int i = tid; i < 64 * 128; i += 256) {
        const int c = i >> 7;
        const int xx = i & 127;
        tile[c * 129 + xx] = x[(((b * CH + c) * HH + y) * WW) + xx];
    }
    __syncthreads();
    for (int i = tid; i < 128 * 64; i += 256) {
        const int xx = i >> 6;
        const int c = i & 63;
        xT[(((b * HH + y) * WW + xx) * CH) + c] = tile[c * 129 + xx];
    }
}

// --------------------------------------------------------------------------
// Kernel 2: build pair-interleaved transposed weights.
//   ck = k*64 + c  (K' index, c fastest inside each tap group of 64)
//   wTp [ck/2][o(64)][2]  from w_deform[o][c][k]           (36864 floats)
//   wToP[ck/2][o(32)][2]  from w_offset[o][c][k], o>=18 -> 0 (18432 floats)
// A-fragment load becomes one coalesced b64 per lane.
// --------------------------------------------------------------------------
__global__ __launch_bounds__(256) void k_prep_weights(const float* __restrict__ w_deform,
                                                      const float* __restrict__ w_offset,
                                                      float* __restrict__ wTp,
                                                      float* __restrict__ wToP) {
    const int stride = gridDim.x * 256;
    for (int i = blockIdx.x * 256 + threadIdx.x; i < 576 * 64; i += stride) {
        const int o = i & 63;
        const int ck = i >> 6;
        const int c = ck & 63;
        const int k = ck >> 6;
        wTp[(ck >> 1) * 128 + (o << 1) + (ck & 1)] = w_deform[(o * CH + c) * KTAP + k];
    }
    for (int i = blockIdx.x * 256 + threadIdx.x; i < 576 * 32; i += stride) {
        const int o = i & 31;
        const int ck = i >> 5;
        const int c = ck & 63;
        const int k = ck >> 6;
        const float v = (o < 18) ? w_offset[(o * CH + c) * KTAP + k] : 0.0f;
        wToP[(ck >> 1) * 64 + (o << 1) + (ck & 1)] = v;
    }
}

// --------------------------------------------------------------------------
// Kernel 3: offset conv via WMMA.  M=32 (rows 18..31 zero), K'=576, N=pixels.
// One wave handles a 16-pixel tile (same b,y; x0..x0+15) and both M-tiles.
// --------------------------------------------------------------------------
__global__ __launch_bounds__(256) void k_offset_conv(const float* __restrict__ xT,
                                                     const float* __restrict__ wToP,
                                                     float* __restrict__ offs) {
    const int tid = threadIdx.x;
    const int wid = tid >> 5;
    const int lane = tid & 31;
    const int tile = blockIdx.x * 8 + wid;   // 8192 tiles total
    const int b = tile >> 10;                // H*(W/16) = 1024 tiles / batch
    const int rem = tile & 1023;
    const int y = rem >> 3;
    const int x0 = (rem & 7) << 4;

    const int n = lane & 15;       // pixel column within tile
    const int half = lane >> 4;    // 0: K elems kk+0,kk+1 ; 1: kk+2,kk+3
    const int dsel = half << 1;
    const int lanelow = lane & 15;

    v8f acc0 = {};
    v8f acc1 = {};

    for (int k = 0; k < KTAP; ++k) {
        const int ky = k / 3;
        const int kx = k % 3;
        const int yy = y + ky - 1;
        const int xx = x0 + n + kx - 1;
        const bool valid = ((unsigned)yy < (unsigned)HH) && ((unsigned)xx < (unsigned)WW);
        const int yyc = valid ? yy : 0;
        const int xxc = valid ? xx : 0;
        const float vm = valid ? 1.0f : 0.0f;
        const int idx = (((b * HH + yyc) * WW + xxc) * CH) + dsel;

        for (int cb = 0; cb < 64; cb += 4) {
            const v2f v = *(const v2f*)(xT + idx + cb);
            v2f bb;
            bb.x = v.x * vm;
            bb.y = v.y * vm;
            const int ckd = k * 64 + cb + dsel;   // even
            const float* wp = wToP + (ckd >> 1) * 64 + (lanelow << 1);
            const v2f aA = *(const v2f*)(wp);        // o = lanelow
            const v2f aB = *(const v2f*)(wp + 32);   // o = 16 + lanelow
            acc0 = WMMA_F32(aA, bb, acc0);
            acc1 = WMMA_F32(aB, bb, acc1);
        }
    }

    // D layout: VGPR r, lanes 0-15 -> M=r ; lanes 16-31 -> M=8+r
    const int px = x0 + n;
    #pragma unroll
    for (int r = 0; r < 8; ++r) {
        const int o = (half ? 8 + r : r);
        offs[(((b * 18 + o) * HH + y) * WW) + px] = acc0[r];
        const int o2 = 16 + o;
        if (o2 < 18) {
            offs[(((b * 18 + o2) * HH + y) * WW) + px] = acc1[r];
        }
    }
}

// --------------------------------------------------------------------------
// Kernel 4: deformable conv via WMMA.  M=64 (4 tiles), K'=576, N=pixels.
// Per-wave: precompute 9 taps x 16 pixels of bilinear data (validity folded
// into weights, corner coords clamped) into LDS; tap data is K-loop invariant
// for 16 consecutive K-steps (ck = k*64 + c ordering).
// --------------------------------------------------------------------------
__global__ __launch_bounds__(256) void k_deform_conv(const float* __restrict__ xT,
                                                     const float* __restrict__ wTp,
                                                     const float* __restrict__ offs,
                                                     const float* __restrict__ bias,
                                                     float* __restrict__ out) {
    __shared__ int   s_idx[8 * 144 * 4];
    __shared__ float s_w[8 * 144 * 4];

    const int tid = threadIdx.x;
    const int wid = tid >> 5;
    const int lane = tid & 31;
    const int tile = blockIdx.x * 8 + wid;
    const int b = tile >> 10;
    const int rem = tile & 1023;
    const int y = rem >> 3;
    const int x0 = (rem & 7) << 4;

    // ---- tap precompute: item i = k*16 + n ----
    for (int i = lane; i < 144; i += 32) {
        const int k = i >> 4;
        const int ni = i & 15;
        const int ky = k / 3;
        const int kx = k % 3;
        const float dy = offs[(((b * 18 + 2 * k) * HH + y) * WW) + x0 + ni];
        const float dx = offs[(((b * 18 + 2 * k + 1) * HH + y) * WW) + x0 + ni];
        const float py = (float)(y - 1 + ky) + dy;
        const float px = (float)(x0 + ni - 1 + kx) + dx;
        const float fy = floorf(py);
        const float fx = floorf(px);
        const float wy = py - fy;
        const float wx = px - fx;
        const int iy0 = (int)fy;
        const int ix0 = (int)fx;
        const int iy1 = iy0 + 1;
        const int ix1 = ix0 + 1;
        const float vy0 = ((unsigned)iy0 < (unsigned)HH) ? 1.0f : 0.0f;
        const float vy1 = ((unsigned)iy1 < (unsigned)HH) ? 1.0f : 0.0f;
        const float vx0 = ((unsigned)ix0 < (unsigned)WW) ? 1.0f : 0.0f;
        const float vx1 = ((unsigned)ix1 < (unsigned)WW) ? 1.0f : 0.0f;
        const int cy0 = min(max(iy0, 0), HH - 1);
        const int cy1 = min(max(iy1, 0), HH - 1);
        const int cx0 = min(max(ix0, 0), WW - 1);
        const int cx1 = min(max(ix1, 0), WW - 1);

        const int base = (wid * 144 + i) << 2;
        s_idx[base + 0] = ((b * HH + cy0) * WW + cx0) * CH;
        s_idx[base + 1] = ((b * HH + cy0) * WW + cx1) * CH;
        s_idx[base + 2] = ((b * HH + cy1) * WW + cx0) * CH;
        s_idx[base + 3] = ((b * HH + cy1) * WW + cx1) * CH;
        s_w[base + 0] = (1.0f - wy) * (1.0f - wx) * vy0 * vx0;
        s_w[base + 1] = (1.0f - wy) * wx * vy0 * vx1;
        s_w[base + 2] = wy * (1.0f - wx) * vy1 * vx0;
        s_w[base + 3] = wy * wx * vy1 * vx1;
    }
    __syncthreads();

    const int n = lane & 15;
    const int half = lane >> 4;
    const int dsel = half << 1;
    const int lanelow = lane & 15;

    v8f acc0 = {};
    v8f acc1 = {};
    v8f acc2 = {};
    v8f acc3 = {};

    for (int k = 0; k < KTAP; ++k) {
        const int base = (wid * 144 + (k << 4) + n) << 2;
        const int i00 = s_idx[base + 0] + dsel;
        const int i01 = s_idx[base + 1] + dsel;
        const int i10 = s_idx[base + 2] + dsel;
        const int i11 = s_idx[base + 3] + dsel;
        const float w00 = s_w[base + 0];
        const float w01 = s_w[base + 1];
        const float w10 = s_w[base + 2];
        const float w11 = s_w[base + 3];

        for (int cb = 0; cb < 64; cb += 4) {
            const v2f v00 = *(const v2f*)(xT + i00 + cb);
            const v2f v01 = *(const v2f*)(xT + i01 + cb);
            const v2f v10 = *(const v2f*)(xT + i10 + cb);
            const v2f v11 = *(const v2f*)(xT + i11 + cb);
            v2f bb;
            bb.x = fmaf(w00, v00.x, fmaf(w01, v01.x, fmaf(w10, v10.x, w11 * v11.x)));
            bb.y = fmaf(w00, v00.y, fmaf(w01, v01.y, fmaf(w10, v10.y, w11 * v11.y)));

            const int ckd = k * 64 + cb + dsel;   // even
            const float* wp = wTp + (ckd >> 1) * 128 + (lanelow << 1);
            const v2f a0 = *(const v2f*)(wp);         // o = lanelow
            const v2f a1 = *(const v2f*)(wp + 32);    // o = 16 + lanelow
            const v2f a2 = *(const v2f*)(wp + 64);    // o = 32 + lanelow
            const v2f a3 = *(const v2f*)(wp + 96);    // o = 48 + lanelow
            acc0 = WMMA_F32(a0, bb, acc0);
            acc1 = WMMA_F32(a1, bb, acc1);
            acc2 = WMMA_F32(a2, bb, acc2);
            acc3 = WMMA_F32(a3, bb, acc3);
        }
    }

    const int px = x0 + n;
    #pragma unroll
    for (int r = 0; r < 8; ++r) {
        const int row = (half ? 8 + r : r);
        const long long base = (((long long)b * OCH) * HH + y) * WW + px;
        out[base + (long long)(row) * HH * WW]      = acc0[r] + bias[row];
        out[base + (long long)(16 + row) * HH * WW] = acc1[r] + bias[16 + row];
        out[base + (long long)(32 + row) * HH * WW] = acc2[r] + bias[32 + row];
        out[base + (long long)(48 + row) * HH * WW] = acc3[r] + bias[48 + row];
    }
}

// --------------------------------------------------------------------------
// Host-side launch.  Workspace layout (floats):
//   xT    : 8*128*128*64 = 8,388,608
//   wTp   : 576*64       = 36,864
//   wToP  : 576*32       = 18,432
//   offs  : 8*18*128*128 = 2,359,296
//   total ~= 43.2 MB
// --------------------------------------------------------------------------
extern "C" void kernel_launch(void* const* d_in, const int* in_sizes, int n_in,
                              void* d_out, int out_size, void* d_ws, size_t ws_size,
                              hipStream_t stream) {
    const float* x        = (const float*)d_in[0];
    const float* w_offset = (const float*)d_in[1];
    const float* w_deform = (const float*)d_in[2];
    const float* b_deform = (const float*)d_in[3];
    float* out = (float*)d_out;

    float* xT   = (float*)d_ws;
    float* wTp  = xT + (size_t)BATCH * HH * WW * CH;
    float* wToP = wTp + 576 * 64;
    float* offs = wToP + 576 * 32;

    k_transpose_x<<<BATCH * HH, 256, 0, stream>>>(x, xT);
    k_prep_weights<<<64, 256, 0, stream>>>(w_deform, w_offset, wTp, wToP);
    k_offset_conv<<<1024, 256, 0, stream>>>(xT, wToP, offs);
    k_deform_conv<<<1024, 256, 0, stream>>>(xT, wTp, offs, b_deform, out);
}